// MinConv2d_77695958384786
// MI455X (gfx1250) — compile-verified
//
#include <hip/hip_runtime.h>
#include <hip/hip_bf16.h>

// 3x3 sliding-window MIN, stride 1, pad 1 (zero padding PARTICIPATES in min).
// Input/Output: (8, 64, 512, 512) fp32. Pure bandwidth problem:
// ~1 GiB HBM traffic -> ~46us floor at 23.3 TB/s on MI455X. Strategy:
// back-to-back async global->LDS b128 staging (CDNA5 ASYNCcnt path),
// separable min (row-min then column-min) from LDS registers-rolling,
// nontemporal output stores. WMMA is deliberately absent: min is not a
// mul-add and any matmul lowering would only add traffic.

#define W 512
#define H 512
#define ROWS 16                 // output rows per block
#define TILE_ROWS (ROWS + 2)    // +1 halo row top & bottom
#define BLOCK 256

static_assert(H % ROWS == 0, "tile must divide height");

__device__ __forceinline__ void async_load_f4(const float* src, float* ldsPtr) {
    // Flat LDS pointer truncated to 32 bits == LDS byte offset (ISA 10.2:
    // LDS aperture maps addr[31:0] directly to the LDS address).
    const unsigned ldsAddr = (unsigned)(unsigned long long)(void*)ldsPtr;
    asm volatile("global_load_async_to_lds_b128 %0, %1, off"
                 :: "v"(ldsAddr), "v"(src)
                 : "memory");
}

__global__ __launch_bounds__(BLOCK)
void MinConv2d_kernel(const float* __restrict__ x, float* __restrict__ out) {
    __shared__ float tile[TILE_ROWS * W];   // 18*512*4 = 36 KB

    const int tid           = threadIdx.x;
    const int tilesPerPlane = H / ROWS;                 // 32
    const int plane         = blockIdx.x / tilesPerPlane;
    const int tileIdx       = blockIdx.x % tilesPerPlane;
    const int r0            = tileIdx * ROWS;           // first output row

    const float* planeBase = x + (size_t)plane * (size_t)(H * W);

    // ---- Stage 16 interior rows (always in range): branch-free,
    //      8 back-to-back async b128 issues per thread. ----
    #pragma unroll
    for (int k = 0; k < (ROWS * (W / 4)) / BLOCK; ++k) {   // 8 iterations
        const int q   = tid + k * BLOCK;    // 0..2047
        const int ri  = q >> 7;             // 0..15  (W/4 == 128 granules/row)
        const int c4  = q & 127;
        const float* src = planeBase + (size_t)(r0 + ri) * W + c4 * 4;
        async_load_f4(src, &tile[(ri + 1) * W + c4 * 4]);
    }

    // ---- Halo rows: waves 0-3 -> top (tile row 0, g = r0-1),
    //      waves 4-7 -> bottom (tile row 17, g = r0+16).
    //      Valid/zero decision is block-uniform per half. ----
    {
        const int  c4   = tid & 127;
        const bool top  = (tid < 128);
        const int  i    = top ? 0 : (TILE_ROWS - 1);
        const int  g    = top ? (r0 - 1) : (r0 + ROWS);
        float* ldsPtr   = &tile[i * W + c4 * 4];
        if (g >= 0 && g < H) {
            async_load_f4(planeBase + (size_t)g * W + c4 * 4, ldsPtr);
        } else {
            // zero-padded rows participate in the min
            *(float4*)ldsPtr = make_float4(0.f, 0.f, 0.f, 0.f);
        }
    }

    asm volatile("s_wait_asynccnt 0x0" ::: "memory");
    __syncthreads();

    // ---- Separable 3x3 min from LDS ----
    float* outPlane = out + (size_t)plane * (size_t)(H * W);

    #pragma unroll
    for (int half = 0; half < 2; ++half) {
        const int c = tid + half * BLOCK;   // column in [0, 512)

        // horizontal 3-min of tile row i at column c (zero pad at edges)
        auto hmin = [&](int i) -> float {
            const float* row = &tile[i * W];
            const float l = (c == 0)     ? 0.0f : row[c - 1];
            const float m = row[c];
            const float r = (c == W - 1) ? 0.0f : row[c + 1];
            return fminf(fminf(l, m), r);
        };

        float h0 = hmin(0);
        float h1 = hmin(1);
        #pragma unroll
        for (int r = 0; r < ROWS; ++r) {
            const float h2 = hmin(r + 2);
            const float v  = fminf(fminf(h0, h1), h2);
            __builtin_nontemporal_store(v, &outPlane[(size_t)(r0 + r) * W + c]);
            h0 = h1;
            h1 = h2;
        }
    }
}

extern "C" void kernel_launch(void* const* d_in, const int* in_sizes, int n_in,
                              void* d_out, int out_size, void* d_ws, size_t ws_size,
                              hipStream_t stream) {
    const float* x  = (const float*)d_in[0];
    float* out      = (float*)d_out;

    const int planes = 8 * 64;                      // N*C
    const int tilesPerPlane = H / ROWS;             // 32
    dim3 grid(planes * tilesPerPlane);              // 16384 blocks
    dim3 block(BLOCK);
    MinConv2d_kernel<<<grid, block, 0, stream>>>(x, out);
}